// GConv_57801669870143
// MI455X (gfx1250) — compile-verified
//
#include <hip/hip_runtime.h>
#include <hip/hip_bf16.h>
#include <math.h>

// Problem constants (from reference)
#define FEAT   128      // in_features
#define OUTF   128      // out_features
#define KDIM   256      // J * F
#define BN_EPS 1e-5f

typedef float v2f __attribute__((ext_vector_type(2)));
typedef float v8f __attribute__((ext_vector_type(8)));

__device__ __forceinline__ void atomic_add_f32(float* p, float v) {
  __hip_atomic_fetch_add(p, v, __ATOMIC_RELAXED, __HIP_MEMORY_SCOPE_AGENT);
}

// ---------------- zero workspace (h accumulator + column stats) ----------------
__global__ void zero_kernel(float* __restrict__ p, long n) {
  long i = (long)blockIdx.x * blockDim.x + threadIdx.x;
  long stride = (long)gridDim.x * blockDim.x;
  for (; i < n; i += stride) p[i] = 0.0f;
}

// ---------------- SpMM scatter: one wave32 per edge, float4 per lane ----------------
__global__ __launch_bounds__(256) void spmm_kernel(
    const float* __restrict__ x, const int* __restrict__ rows,
    const int* __restrict__ cols, const float* __restrict__ vals,
    float* __restrict__ h, int nEdges, int colOff) {
  int tid = blockIdx.x * blockDim.x + threadIdx.x;
  int e = tid >> 5;                 // wave index == edge index
  if (e >= nEdges) return;
  int lane = tid & 31;
  int r = rows[e];
  int c = cols[e];
  float v = vals[e];
  int f = lane << 2;                // 32 lanes * 4 feats = 128
  float4 xf = *reinterpret_cast<const float4*>(x + (size_t)c * FEAT + f);
  float* dst = h + (size_t)r * KDIM + colOff + f;
  atomic_add_f32(dst + 0, v * xf.x);
  atomic_add_f32(dst + 1, v * xf.y);
  atomic_add_f32(dst + 2, v * xf.z);
  atomic_add_f32(dst + 3, v * xf.w);
}

// ---------------- GEMM y = h @ W^T + b via v_wmma_f32_16x16x4_f32 ----------------
// Block: 256 threads = 8 waves; block owns 16 rows of h, wave w owns output
// columns [16w, 16w+16). K loop: 256 in steps of 4 -> 64 chained WMMAs.
#define LDSP 258   // padded row stride (floats) to avoid LDS bank conflicts
__global__ __launch_bounds__(256) void gemm_wmma_kernel(
    const float* __restrict__ h, const float* __restrict__ W,
    const float* __restrict__ bias, float* __restrict__ y) {
  __shared__ float ah[16 * LDSP];
  const int tid = threadIdx.x;
  const int m0 = blockIdx.x * 16;

  // Cooperative, fully coalesced load of the 16x256 fp32 A tile into LDS.
  #pragma unroll
  for (int i = 0; i < 16; ++i)
    ah[i * LDSP + tid] = h[(size_t)(m0 + i) * KDIM + tid];
  __syncthreads();

  const int wave = tid >> 5;
  const int lane = tid & 31;
  const int lr   = lane & 15;       // row (A) / col (B,D) selector within tile
  const int lhi  = lane >> 4;       // half-wave selector
  const int col  = wave * 16 + lr;  // global output column for this lane

  // ISA 7.12.2: 32-bit A 16x4 -> lanes 0-15 hold {K,K+1}, lanes 16-31 hold {K+2,K+3}
  //             B 4x16 mirrored: lane holds B[kk][col], B[kk+1][col]
  v8f acc = {};
  const float* wrow = W + (size_t)col * KDIM;   // B[k][col] = W_fc[col*KDIM + k]
  #pragma unroll 8
  for (int k = 0; k < KDIM; k += 4) {
    const int kk = k + 2 * lhi;
    v2f a = *reinterpret_cast<const v2f*>(&ah[lr * LDSP + kk]);  // 8B ds_load
    v2f b = *reinterpret_cast<const v2f*>(wrow + kk);            // 8B global load (L2-resident)
    acc = __builtin_amdgcn_wmma_f32_16x16x4_f32(
        /*neg_a=*/false, a, /*neg_b=*/false, b,
        /*c_mod=*/(short)0, acc, /*reuse_a=*/false, /*reuse_b=*/false);
  }

  // D layout (32-bit C/D 16x16): VGPR r -> M = r (lanes 0-15) / M = r+8 (lanes 16-31)
  const float bc = bias[col];
  #pragma unroll
  for (int r = 0; r < 8; ++r) {
    int row = m0 + r + 8 * lhi;
    y[(size_t)row * OUTF + col] = acc[r] + bc;
  }
}

// ---------------- per-column sum / sum-of-squares ----------------
__global__ __launch_bounds__(128) void colstats_kernel(
    const float* __restrict__ y, float* __restrict__ stats, int n) {
  int c = threadIdx.x;              // 128 threads = 128 columns
  float s = 0.f, s2 = 0.f;
  for (int r = blockIdx.x; r < n; r += gridDim.x) {
    float v = y[(size_t)r * OUTF + c];
    s += v; s2 += v * v;
  }
  atomic_add_f32(&stats[c], s);
  atomic_add_f32(&stats[OUTF + c], s2);
}

// ---------------- BatchNorm apply (training mode, biased variance) ----------------
__global__ void bn_kernel(float* __restrict__ y, const float* __restrict__ stats,
                          const float* __restrict__ gamma, const float* __restrict__ beta,
                          int n) {
  long idx = (long)blockIdx.x * blockDim.x + threadIdx.x;
  long total = (long)n * OUTF;
  if (idx >= total) return;
  int c = (int)(idx & (OUTF - 1));
  float invN  = 1.0f / (float)n;
  float mean  = stats[c] * invN;
  float var   = stats[OUTF + c] * invN - mean * mean;   // E[y^2] - mean^2 (biased)
  float scale = rsqrtf(var + BN_EPS) * gamma[c];
  y[idx] = (y[idx] - mean) * scale + beta[c];
}

extern "C" void kernel_launch(void* const* d_in, const int* in_sizes, int n_in,
                              void* d_out, int out_size, void* d_ws, size_t ws_size,
                              hipStream_t stream) {
  const float* x     = (const float*)d_in[0];
  const int*   rows0 = (const int*)  d_in[1];
  const int*   cols0 = (const int*)  d_in[2];
  const float* vals0 = (const float*)d_in[3];
  const int*   rows1 = (const int*)  d_in[4];
  const int*   cols1 = (const int*)  d_in[5];
  const float* vals1 = (const float*)d_in[6];
  const float* W     = (const float*)d_in[7];   // [OUT, J*F] row-major
  const float* bfc   = (const float*)d_in[8];
  const float* gamma = (const float*)d_in[9];
  const float* beta  = (const float*)d_in[10];
  float* y = (float*)d_out;

  const int n  = in_sizes[0] / FEAT;   // 50000 nodes (multiple of 16)
  const int nE = in_sizes[1];          // 600000 edges per adjacency

  // Workspace layout: h[n*KDIM] | colsum[128] | colsumsq[128]
  float* h     = (float*)d_ws;
  float* stats = h + (size_t)n * KDIM;

  const long zn = (long)n * KDIM + 2 * OUTF;
  zero_kernel<<<2048, 256, 0, stream>>>(h, zn);

  const int spmmBlocks = (int)(((long)nE * 32 + 255) / 256);
  spmm_kernel<<<spmmBlocks, 256, 0, stream>>>(x, rows0, cols0, vals0, h, nE, 0);
  spmm_kernel<<<spmmBlocks, 256, 0, stream>>>(x, rows1, cols1, vals1, h, nE, FEAT);

  gemm_wmma_kernel<<<n / 16, 256, 0, stream>>>(h, W, bfc, y);

  colstats_kernel<<<512, 128, 0, stream>>>(y, stats, n);

  const long total = (long)n * OUTF;
  bn_kernel<<<(int)((total + 255) / 256), 256, 0, stream>>>(y, stats, gamma, beta, n);
}